// SpecAugment_15728170237988
// MI455X (gfx1250) — compile-verified
//
#include <hip/hip_runtime.h>
#include <stdint.h>

// ---------------------------------------------------------------------------
// SpecAugment for MI455X (gfx1250).
//   specaug_apply (emitted first so disasm snippet shows the TDM path):
//     stream each 12KB mel row via Tensor Data Mover (tensor_load_to_lds),
//     s_wait_tensorcnt, ds_load_b128 from LDS, analytic mask, NT b128 stores.
//   specaug_mask_params: replicate JAX threefry PRNG (partitionable mode) to
//     produce per-batch (width,start) pairs: 2 freq + 2 time masks.
// Memory-bound: ~197MB traffic -> ~8.4us HBM floor; NT stores keep the 98MB
// input resident in the 192MB L2 across graph replays.
// ---------------------------------------------------------------------------

typedef float        f32x4    __attribute__((ext_vector_type(4)));
typedef unsigned int uint32x4 __attribute__((ext_vector_type(4)));
typedef int          int32x4  __attribute__((ext_vector_type(4)));
typedef int          int32x8  __attribute__((ext_vector_type(8)));

constexpr int B_    = 64;
constexpr int NMEL  = 128;
constexpr int TLEN  = 3000;
constexpr int FREQ_MAX = 20;   // FREQ_MASK_MAX
constexpr int TIME_MAX = 40;   // TIME_MASK_MAX
constexpr int ROWS  = B_ * NMEL;           // 8192 rows (C==1)
constexpr int WAVES_PER_BLOCK = 4;         // 128 threads/block, wave32
constexpr int ROW_F4   = TLEN / 4;         // 750 float4 per row
constexpr int ROW_BYTES = TLEN * 4;        // 12000 bytes per row

// --------------------------- TDM row streaming -----------------------------
// Build a 1-row 2D tile descriptor (D#) and issue TENSOR_LOAD_TO_LDS.
// Group0: [count=1][lds_addr][global_addr(57b)][type=2]
// Group1: data_size=4B, tensor_dim0=tile_dim0=TLEN, tensor_dim1=tile_dim1=1.
// Groups 2/3 unused (2D tile) -> zeros. 6-arg builtin form (clang-23 lane).
__device__ __forceinline__ void tdm_load_row(const float* src, uint32_t lds_off) {
  const uint64_t ga = (uint64_t)(uintptr_t)src;
  uint32x4 g0;
  g0[0] = 1u;                                   // count=1, user-mode, no gather
  g0[1] = lds_off;                              // LDS byte address (dyn LDS base 0)
  g0[2] = (uint32_t)ga;                         // global_addr[31:0]
  g0[3] = (uint32_t)((ga >> 32) & 0x1FFFFFFull) | (2u << 30);  // addr[56:32]|type=2
  int32x8 g1;
  g1[0] = (int)(2u << 16);                      // wg_mask=0, data_size=2 (4B)
  g1[1] = (int)(((uint32_t)TLEN & 0xFFFFu) << 16);   // tensor_dim0 lo16
  g1[2] = (int)(((uint32_t)TLEN >> 16) | (1u << 16));// tensor_dim0 hi | tensor_dim1 lo=1
  g1[3] = (int)((uint32_t)TLEN << 16);          // tensor_dim1 hi=0 | tile_dim0=TLEN
  g1[4] = 1;                                    // tile_dim1=1, tile_dim2=0
  g1[5] = TLEN;                                 // tensor_dim0_stride lo32
  g1[6] = 0;
  g1[7] = 0;
  int32x4 gz4  = {0, 0, 0, 0};
  int32x8 gz8  = {0, 0, 0, 0, 0, 0, 0, 0};
  __builtin_amdgcn_tensor_load_to_lds(g0, g1, gz4, gz4, gz8, 0);
}

__global__ void __launch_bounds__(32 * WAVES_PER_BLOCK)
specaug_apply(const float* __restrict__ mel, const int* __restrict__ params,
              float* __restrict__ out) {
  extern __shared__ float smem[];  // WAVES_PER_BLOCK * 12000 bytes, base off 0

  // Wave-private row: no cross-wave barriers needed anywhere.
  const int wave = __builtin_amdgcn_readfirstlane((int)(threadIdx.x >> 5));
  const int lane = (int)(threadIdx.x & 31);
  const int row  = (int)blockIdx.x * WAVES_PER_BLOCK + wave;   // 0..8191
  const int b    = row >> 7;            // NMEL == 128
  const int m    = row & (NMEL - 1);

  const int* p = params + b * 8;
  const int fw0 = p[0], fs0 = p[1], fw1 = p[2], fs1 = p[3];
  const int tw0 = p[4], ts0 = p[5], tw1 = p[6], ts1 = p[7];
  const bool fdrop = ((unsigned)(m - fs0) < (unsigned)fw0) ||
                     ((unsigned)(m - fs1) < (unsigned)fw1);

  const float* src = mel + (size_t)row * TLEN;
  float*       dst = out + (size_t)row * TLEN;

  // Async DMA: whole 12KB row -> this wave's LDS slice, then wait TENSORcnt.
  tdm_load_row(src, (uint32_t)(wave * ROW_BYTES));
  __builtin_amdgcn_s_wait_tensorcnt(0);

  const f32x4* sm4 = (const f32x4*)smem + wave * ROW_F4;
  f32x4* d4 = (f32x4*)dst;

  for (int i = lane; i < ROW_F4; i += 32) {
    const f32x4 v = sm4[i];                 // ds_load_b128
    const int t = i * 4;
    f32x4 r;
    r.x = (fdrop || ((unsigned)(t + 0 - ts0) < (unsigned)tw0)
                 || ((unsigned)(t + 0 - ts1) < (unsigned)tw1)) ? 0.0f : v.x;
    r.y = (fdrop || ((unsigned)(t + 1 - ts0) < (unsigned)tw0)
                 || ((unsigned)(t + 1 - ts1) < (unsigned)tw1)) ? 0.0f : v.y;
    r.z = (fdrop || ((unsigned)(t + 2 - ts0) < (unsigned)tw0)
                 || ((unsigned)(t + 2 - ts1) < (unsigned)tw1)) ? 0.0f : v.z;
    r.w = (fdrop || ((unsigned)(t + 3 - ts0) < (unsigned)tw0)
                 || ((unsigned)(t + 3 - ts1) < (unsigned)tw1)) ? 0.0f : v.w;
    // NT store: don't let the 98MB output thrash L2 (input stays resident).
    __builtin_nontemporal_store(r, d4 + i);
  }
}

// ----------------------------- Threefry-2x32 -------------------------------
__device__ __forceinline__ void tf2x32(uint32_t k0, uint32_t k1,
                                       uint32_t x0, uint32_t x1,
                                       uint32_t& y0, uint32_t& y1) {
  const uint32_t ks2 = k0 ^ k1 ^ 0x1BD11BDAu;
  x0 += k0; x1 += k1;
#define TFR(r) { x0 += x1; x1 = (x1 << (r)) | (x1 >> (32 - (r))); x1 ^= x0; }
  TFR(13) TFR(15) TFR(26) TFR(6)   x0 += k1;  x1 += ks2 + 1u;
  TFR(17) TFR(29) TFR(16) TFR(24)  x0 += ks2; x1 += k0 + 2u;
  TFR(13) TFR(15) TFR(26) TFR(6)   x0 += k0;  x1 += k1 + 3u;
  TFR(17) TFR(29) TFR(16) TFR(24)  x0 += k1;  x1 += ks2 + 4u;
  TFR(13) TFR(15) TFR(26) TFR(6)   x0 += ks2; x1 += k0 + 5u;
#undef TFR
  y0 = x0; y1 = x1;
}

// 32-bit random_bits in partitionable mode: xor-fold of the two output words,
// counter = (hi=0, lo=i) for flat index i.
__device__ __forceinline__ uint32_t tf_fold(uint32_t k0, uint32_t k1, uint32_t i) {
  uint32_t a, b; tf2x32(k0, k1, 0u, i, a, b); return a ^ b;
}

// One dimension of _sample_masks for batch b: writes w0,s0,w1,s1.
__device__ void sample_dim(uint32_t k0, uint32_t k1, uint32_t span, int D,
                           int b, int* out) {
  uint32_t kw0, kw1, ks0, ks1, k10, k11, k20, k21;
  tf2x32(k0, k1, 0u, 0u, kw0, kw1);   // kw = split(key)[0]
  tf2x32(k0, k1, 0u, 1u, ks0, ks1);   // ks = split(key)[1]
  tf2x32(kw0, kw1, 0u, 0u, k10, k11); // randint internal split -> k1
  tf2x32(kw0, kw1, 0u, 1u, k20, k21); // randint internal split -> k2
  const uint32_t m0   = 65536u % span;
  const uint32_t mult = (m0 * m0) % span;   // (2^32 mod span); == 16 for 20 & 40
  for (int m = 0; m < 2; ++m) {
    const uint32_t i = (uint32_t)(b * 2 + m);
    const uint32_t h = tf_fold(k10, k11, i);
    const uint32_t l = tf_fold(k20, k21, i);
    const uint32_t w = ((h % span) * mult + (l % span)) % span;
    const uint32_t ub = tf_fold(ks0, ks1, i);
    float u = __uint_as_float((ub >> 9) | 0x3F800000u) - 1.0f;  // [0,1)
    u = fmaxf(u, 0.0f);
    const int st = (int)(u * (float)(D - (int)w));  // f32 mul, trunc cast
    out[2 * m + 0] = (int)w;
    out[2 * m + 1] = st;
  }
}

__global__ void specaug_mask_params(int* __restrict__ params) {
  const int b = (int)(blockIdx.x * blockDim.x + threadIdx.x);
  if (b >= B_) return;
  // root key = threefry_seed(42) = (0,42); kf,kt = split(root)
  uint32_t kf0, kf1, kt0, kt1;
  tf2x32(0u, 42u, 0u, 0u, kf0, kf1);
  tf2x32(0u, 42u, 0u, 1u, kt0, kt1);
  sample_dim(kf0, kf1, (uint32_t)FREQ_MAX, NMEL, b, params + b * 8);
  sample_dim(kt0, kt1, (uint32_t)TIME_MAX, TLEN, b, params + b * 8 + 4);
}

// ---------------------------------------------------------------------------
extern "C" void kernel_launch(void* const* d_in, const int* in_sizes, int n_in,
                              void* d_out, int out_size, void* d_ws, size_t ws_size,
                              hipStream_t stream) {
  (void)in_sizes; (void)n_in; (void)out_size; (void)ws_size;
  const float* mel = (const float*)d_in[0];
  float* out = (float*)d_out;
  int* params = (int*)d_ws;   // 64 batches * 8 ints = 2KB scratch

  specaug_mask_params<<<1, 64, 0, stream>>>(params);

  const int blocks = ROWS / WAVES_PER_BLOCK;                  // 2048
  const size_t dyn_lds = (size_t)WAVES_PER_BLOCK * ROW_BYTES; // 48000 bytes
  specaug_apply<<<blocks, 32 * WAVES_PER_BLOCK, dyn_lds, stream>>>(mel, params, out);
}